// ThreeBlindMice_73873437491383
// MI455X (gfx1250) — compile-verified
//
#include <hip/hip_runtime.h>
#include <math.h>

#define N_SYM  2048
#define HIDDEN 2048
#define IOTA_F (1.0f/137.0f)
#define PHI_F  0.618033988749894f
#define KC 32
#define LSTRIDE 36   // LDS row stride (floats): mult of 4 (16B alignment), bank-conflict free

typedef float v2f __attribute__((ext_vector_type(2)));
typedef float v8f __attribute__((ext_vector_type(8)));

__device__ __forceinline__ float sigm(float x) { return 1.0f / (1.0f + expf(-x)); }

// CDNA5 async memory->LDS copy (16B per lane), tracked by ASYNCcnt.
// LDS byte address = low 32 bits of the generic pointer.
__device__ __forceinline__ void async_copy_b128(const float* gptr, const float* lptr) {
  asm volatile("global_load_async_to_lds_b128 %0, %1, off"
               :: "v"((unsigned int)(unsigned long long)lptr),
                  "v"((unsigned long long)gptr)
               : "memory");
}
__device__ __forceinline__ void wait_async0() {
  asm volatile("s_wait_asynccnt 0x0" ::: "memory");
}

// ---------------------------------------------------------------------------
// Phase A: instance-norm(5) -> rfft(5) -> sigmoid -> softmax(3) -> bilinear ->
//          u1 mask  ==> B0 [16][2048] (rows 9..15 zero-padded for WMMA M=16)
// ---------------------------------------------------------------------------
__global__ void __launch_bounds__(256)
phaseA_kernel(const float* __restrict__ signal, const float* __restrict__ u1,
              const float* __restrict__ bilW, const float* __restrict__ bilb,
              float* __restrict__ B0) {
  int n = blockIdx.x * blockDim.x + threadIdx.x;
  if (n >= N_SYM) return;
  float x[5];
  float mean = 0.0f;
  #pragma unroll
  for (int t = 0; t < 5; t++) { x[t] = signal[t * N_SYM + n]; mean += x[t]; }
  mean *= 0.2f;
  float var = 0.0f;
  #pragma unroll
  for (int t = 0; t < 5; t++) { float d = x[t] - mean; var += d * d; }
  var *= 0.2f;
  float rstd = rsqrtf(var + 1e-9f);
  #pragma unroll
  for (int t = 0; t < 5; t++) x[t] = (x[t] - mean) * rstd;

  // rfft over 5 points, k = 0,1,2 ; b_k = Re * Im with nan_to_num
  float br[3];
  const float TW = 1.2566370614359172954f;  // 2*pi/5
  #pragma unroll
  for (int k = 0; k < 3; k++) {
    float re = 0.0f, im = 0.0f;
    #pragma unroll
    for (int t = 0; t < 5; t++) {
      float a = TW * (float)(k * t);
      re += x[t] * cosf(a);
      im -= x[t] * sinf(a);
    }
    float p = re * im;
    if (isnan(p))      p = IOTA_F;
    else if (isinf(p)) p = (p > 0.0f) ? 1.0f : IOTA_F;
    br[k] = p;
  }
  // sigmoid then softmax over the 3 bins
  float sb[3]; float mx = -1e30f;
  #pragma unroll
  for (int k = 0; k < 3; k++) { sb[k] = sigm(br[k]); mx = fmaxf(mx, sb[k]); }
  float se = 0.0f;
  #pragma unroll
  for (int k = 0; k < 3; k++) { sb[k] = expf(sb[k] - mx); se += sb[k]; }
  float inv = 1.0f / se;
  float p3[3];
  #pragma unroll
  for (int k = 0; k < 3; k++) p3[k] = sb[k] * inv;

  // bilinear 3 -> 9, then stochastic mask, write transposed [9][2048]
  #pragma unroll
  for (int kk = 0; kk < 9; kk++) {
    float acc = bilb[kk];
    #pragma unroll
    for (int i = 0; i < 3; i++)
      #pragma unroll
      for (int j = 0; j < 3; j++)
        acc += p3[i] * bilW[kk * 9 + i * 3 + j] * p3[j];
    float m = (u1[kk * N_SYM + n] < PHI_F) ? 1.0f : 0.0f;
    B0[kk * N_SYM + n] = IOTA_F + acc * m;
  }
  #pragma unroll
  for (int kk = 9; kk < 16; kk++) B0[kk * N_SYM + n] = 0.0f;
}

// ---------------------------------------------------------------------------
// LSTM-cell GEMM (h0=c0=0 => Whh dead, f-gate dead).
// Block: 192 threads = 6 waves, covers 32 hidden cols x gates {i,g,o}.
// Weights: double-buffered async global->LDS copies; V_WMMA_F32_16X16X4_F32.
// Epilogue fuses sigmoid/tanh and optional u-mask; writes H [16][2048].
// ---------------------------------------------------------------------------
__global__ void __launch_bounds__(192)
lstm_gemm_kernel(const float* __restrict__ A,      // [16][2048]
                 const float* __restrict__ W,      // [8192][2048] row-major
                 const float* __restrict__ bih, const float* __restrict__ bhh,
                 const float* __restrict__ u, int use_mask,
                 float* __restrict__ Hout) {       // [16][2048]
  __shared__ float wt[2 * 96 * LSTRIDE];
  const int tid  = threadIdx.x;
  const int lane = tid & 31;
  const int w    = tid >> 5;       // wave 0..5
  const int gg   = w >> 1;         // gate group: 0=i, 1=g, 2=o
  const int ct   = w & 1;          // 16-col tile within the 32-col slab
  const int j0   = blockIdx.x * 32;
  const int half = lane >> 4;
  const int n    = lane & 15;
  const int arow = lane & 15;
  const int ldsrow = gg * 32 + ct * 16 + n;

  auto stage = [&](int k0, int buf) {
    #pragma unroll
    for (int it = 0; it < 4; it++) {
      int i  = tid + it * 192;
      int r  = i >> 3;
      int c4 = (i & 7) << 2;
      int g2 = r >> 5;
      int jr = r & 31;
      int go = (g2 + (g2 > 0 ? 1 : 0)) * HIDDEN;   // gate row offsets: 0, 2H, 3H
      async_copy_b128(W + (size_t)(go + j0 + jr) * 2048 + k0 + c4,
                      &wt[buf * (96 * LSTRIDE) + r * LSTRIDE + c4]);
    }
  };

  stage(0, 0);
  wait_async0();
  __syncthreads();

  v8f acc = {};
  int cur = 0;
  for (int k0 = 0; k0 < 2048; k0 += KC) {
    if (k0 + KC < 2048) stage(k0 + KC, cur ^ 1);   // prefetch next chunk (no wait)

    v2f af[8];
    #pragma unroll
    for (int q = 0; q < 8; q++)
      af[q] = *(const v2f*)(A + arow * 2048 + k0 + q * 4 + 2 * half);

    const float* wb = &wt[cur * (96 * LSTRIDE)];
    #pragma unroll
    for (int q = 0; q < 8; q++) {
      v2f b = *(const v2f*)(&wb[ldsrow * LSTRIDE + q * 4 + 2 * half]);
      acc = __builtin_amdgcn_wmma_f32_16x16x4_f32(false, af[q], false, b,
                                                  (short)0, acc, false, false);
    }
    wait_async0();      // next-buffer copies (from every wave) landed...
    __syncthreads();    // ...block-wide; also: everyone done reading cur
    cur ^= 1;
  }

  // dump gate tiles to LDS: wt[gate*512 + m*32 + col]
  #pragma unroll
  for (int v = 0; v < 8; v++) {
    int m = v + 8 * half;
    wt[gg * 512 + m * 32 + ct * 16 + n] = acc[v];
  }
  __syncthreads();

  for (int idx = tid; idx < 512; idx += 192) {
    int m = idx >> 5, j = idx & 31;
    int col = j0 + j;
    float iv = wt[0 * 512 + m * 32 + j] + bih[col]              + bhh[col];
    float gv = wt[1 * 512 + m * 32 + j] + bih[2 * HIDDEN + col] + bhh[2 * HIDDEN + col];
    float ov = wt[2 * 512 + m * 32 + j] + bih[3 * HIDDEN + col] + bhh[3 * HIDDEN + col];
    float cs = sigm(iv) * tanhf(gv);
    float h  = sigm(ov) * tanhf(cs);
    if (use_mask && m < 9)
      h = IOTA_F + h * ((u[m * HIDDEN + col] < PHI_F) ? 1.0f : 0.0f);
    Hout[m * HIDDEN + col] = h;
  }
}

// ---------------------------------------------------------------------------
// Final linear: [16,2048] x lin_W.T -> raw [9][10240] (rows >= 9 dropped).
// Block: 128 threads = 4 waves, 64 output cols. Same async double-buffering.
// ---------------------------------------------------------------------------
__global__ void __launch_bounds__(128)
gemm_lin_kernel(const float* __restrict__ A,      // [16][2048]
                const float* __restrict__ W,      // [10240][2048]
                const float* __restrict__ bias,   // [10240]
                float* __restrict__ Out) {        // [9][10240] flat
  __shared__ float wt[2 * 64 * LSTRIDE];
  const int tid  = threadIdx.x;
  const int lane = tid & 31;
  const int w    = tid >> 5;       // 0..3
  const int c0   = blockIdx.x * 64;
  const int half = lane >> 4;
  const int n    = lane & 15;
  const int arow = lane & 15;
  const int ldsrow = w * 16 + n;

  auto stage = [&](int k0, int buf) {
    #pragma unroll
    for (int it = 0; it < 4; it++) {
      int i  = tid + it * 128;
      int r  = i >> 3;
      int c4 = (i & 7) << 2;
      async_copy_b128(W + (size_t)(c0 + r) * 2048 + k0 + c4,
                      &wt[buf * (64 * LSTRIDE) + r * LSTRIDE + c4]);
    }
  };

  stage(0, 0);
  wait_async0();
  __syncthreads();

  v8f acc = {};
  int cur = 0;
  for (int k0 = 0; k0 < 2048; k0 += KC) {
    if (k0 + KC < 2048) stage(k0 + KC, cur ^ 1);

    v2f af[8];
    #pragma unroll
    for (int q = 0; q < 8; q++)
      af[q] = *(const v2f*)(A + arow * 2048 + k0 + q * 4 + 2 * half);

    const float* wb = &wt[cur * (64 * LSTRIDE)];
    #pragma unroll
    for (int q = 0; q < 8; q++) {
      v2f b = *(const v2f*)(&wb[ldsrow * LSTRIDE + q * 4 + 2 * half]);
      acc = __builtin_amdgcn_wmma_f32_16x16x4_f32(false, af[q], false, b,
                                                  (short)0, acc, false, false);
    }
    wait_async0();
    __syncthreads();
    cur ^= 1;
  }

  int col = c0 + w * 16 + n;
  float bv = bias[col];
  #pragma unroll
  for (int v = 0; v < 8; v++) {
    int m = v + 8 * half;
    if (m < 9) Out[(size_t)m * 10240 + col] = acc[v] + bv;
  }
}

// ---------------------------------------------------------------------------
// Finale: flat-reshape 9-sum, instance-norm over 2048, softmax over 2048.
// One block per sample (5 blocks).
// ---------------------------------------------------------------------------
__global__ void __launch_bounds__(256)
finale_kernel(const float* __restrict__ G, float* __restrict__ out) {
  __shared__ float tl[2048];
  __shared__ float red[256];
  __shared__ float red2[256];
  const int s = blockIdx.x, tid = threadIdx.x;
  size_t base = (size_t)s * 18432;

  float ps = 0.0f, ps2 = 0.0f;
  for (int nn = tid; nn < 2048; nn += 256) {
    float sum = 0.0f;
    size_t b2 = base + (size_t)nn * 9;
    #pragma unroll
    for (int k = 0; k < 9; k++) sum += G[b2 + k];
    tl[nn] = sum; ps += sum; ps2 += sum * sum;
  }
  red[tid] = ps; red2[tid] = ps2; __syncthreads();
  for (int st = 128; st > 0; st >>= 1) {
    if (tid < st) { red[tid] += red[tid + st]; red2[tid] += red2[tid + st]; }
    __syncthreads();
  }
  float mean = red[0] * (1.0f / 2048.0f);
  float var  = red2[0] * (1.0f / 2048.0f) - mean * mean;
  float rstd = rsqrtf(var + 1e-9f);
  __syncthreads();

  float lmax = -1e30f;
  for (int nn = tid; nn < 2048; nn += 256) {
    float z = (tl[nn] - mean) * rstd; tl[nn] = z; lmax = fmaxf(lmax, z);
  }
  __syncthreads();
  red[tid] = lmax; __syncthreads();
  for (int st = 128; st > 0; st >>= 1) {
    if (tid < st) red[tid] = fmaxf(red[tid], red[tid + st]);
    __syncthreads();
  }
  float zmax = red[0];
  __syncthreads();

  float lsum = 0.0f;
  for (int nn = tid; nn < 2048; nn += 256) {
    float e = expf(tl[nn] - zmax); tl[nn] = e; lsum += e;
  }
  red[tid] = lsum; __syncthreads();
  for (int st = 128; st > 0; st >>= 1) {
    if (tid < st) red[tid] += red[tid + st];
    __syncthreads();
  }
  float inv = 1.0f / red[0];
  __syncthreads();
  for (int nn = tid; nn < 2048; nn += 256) out[s * 2048 + nn] = tl[nn] * inv;
}

// ---------------------------------------------------------------------------
extern "C" void kernel_launch(void* const* d_in, const int* in_sizes, int n_in,
                              void* d_out, int out_size, void* d_ws, size_t ws_size,
                              hipStream_t stream) {
  (void)in_sizes; (void)n_in; (void)out_size; (void)ws_size;
  const float* signal = (const float*)d_in[0];
  const float* u1     = (const float*)d_in[1];
  const float* u2     = (const float*)d_in[2];
  const float* bilW   = (const float*)d_in[3];
  const float* bilb   = (const float*)d_in[4];
  const float* Wih1   = (const float*)d_in[5];
  // d_in[6] Whh1: dead (h0 = 0)
  const float* bih1   = (const float*)d_in[7];
  const float* bhh1   = (const float*)d_in[8];
  const float* Wih2   = (const float*)d_in[9];
  // d_in[10] Whh2: dead
  const float* bih2   = (const float*)d_in[11];
  const float* bhh2   = (const float*)d_in[12];
  const float* linW   = (const float*)d_in[13];
  const float* linb   = (const float*)d_in[14];

  float* ws = (float*)d_ws;
  float* B0 = ws;                     // 16*2048
  float* H1 = ws + 1 * 16 * 2048;     // 16*2048
  float* H2 = ws + 2 * 16 * 2048;     // 16*2048
  float* G3 = ws + 3 * 16 * 2048;     // 9*10240
  float* out = (float*)d_out;

  phaseA_kernel<<<8, 256, 0, stream>>>(signal, u1, bilW, bilb, B0);
  lstm_gemm_kernel<<<64, 192, 0, stream>>>(B0, Wih1, bih1, bhh1, u2, 1, H1);
  lstm_gemm_kernel<<<64, 192, 0, stream>>>(H1, Wih2, bih2, bhh2, u2, 0, H2);
  gemm_lin_kernel<<<160, 128, 0, stream>>>(H2, linW, linb, G3);
  finale_kernel<<<5, 256, 0, stream>>>(G3, out);
}